// GaussianKANLayer_10436770529857
// MI455X (gfx1250) — compile-verified
//
#include <hip/hip_runtime.h>

// Problem dims (fixed by the reference)
#define BATCH   1024
#define IN_DIM  256
#define N_GAUSS 512
#define OUT_DIM 256

typedef __attribute__((ext_vector_type(2))) float v2f;
typedef __attribute__((ext_vector_type(8))) float v8f;

// Full-precision fp32 WMMA: D(16x16,f32) = A(16x4,f32) * B(4x16,f32) + C
// 8-arg VOP3P form: (neg_a, A, neg_b, B, c_mod, C, reuse_a, reuse_b)
__device__ __forceinline__ v8f wmma_f32(v2f a, v2f b, v8f c) {
    return __builtin_amdgcn_wmma_f32_16x16x4_f32(
        /*neg_a=*/false, a, /*neg_b=*/false, b,
        /*c_mod=*/(short)0, c, /*reuse_a=*/false, /*reuse_b=*/false);
}

// ---------------------------------------------------------------------------
// Prep: per (g,k) compute inv = 1/s^2, tables b1p = -0.5*inv and b2 = c*inv,
// and per-g constant cst[g] = -0.5 * sum_k c^2 * inv. One block per gaussian.
// ---------------------------------------------------------------------------
__global__ void gkan_prep(const float* __restrict__ centers,
                          const float* __restrict__ stds,
                          float* __restrict__ b1p,
                          float* __restrict__ b2,
                          float* __restrict__ cst) {
    __shared__ float red[IN_DIM];
    const int g = blockIdx.x;
    const int k = threadIdx.x;
    const int idx = g * IN_DIM + k;
    const float s = stds[idx];
    const float c = centers[idx];
    const float inv = 1.0f / (s * s);
    b1p[idx] = -0.5f * inv;
    b2[idx]  = c * inv;
    red[k]   = c * c * inv;
    __syncthreads();
    for (int off = IN_DIM / 2; off > 0; off >>= 1) {
        if (k < off) red[k] += red[k + off];
        __syncthreads();
    }
    if (k == 0) cst[g] = -0.5f * red[0];
}

// ---------------------------------------------------------------------------
// Phase 1: E = (x*x) @ b1p^T + x @ b2^T  (two GEMMs fused into one WMMA
// accumulator chain), then G = exp(E + cst[g]). One wave per 16x16 tile.
// M = BATCH, N = N_GAUSS, K = IN_DIM, f32 WMMA K-step = 4.
// ---------------------------------------------------------------------------
__global__ void __launch_bounds__(128)
gkan_gauss(const float* __restrict__ x,
           const float* __restrict__ b1p,
           const float* __restrict__ b2,
           const float* __restrict__ cst,
           float* __restrict__ G) {
    const int wave  = (blockIdx.x * blockDim.x + threadIdx.x) >> 5;
    const int lane  = threadIdx.x & 31;
    const int gn    = wave & ((N_GAUSS / 16) - 1);  // gauss tile (0..31)
    const int bm    = wave >> 5;                    // batch tile (0..63)
    const int row   = lane & 15;                    // M index (A) == N index (B)
    const int khalf = lane >> 4;                    // lanes 16-31 hold K+2

    const float* xrow  = x   + (bm * 16 + row) * IN_DIM;
    const float* b1row = b1p + (gn * 16 + row) * IN_DIM;  // B col == lane&15
    const float* b2row = b2  + (gn * 16 + row) * IN_DIM;

    v8f acc = {};
#pragma unroll 4
    for (int k0 = 0; k0 < IN_DIM; k0 += 4) {
        const int kk = k0 + 2 * khalf;               // even -> 8B aligned
        const v2f a   = *(const v2f*)(xrow  + kk);
        const v2f bb1 = *(const v2f*)(b1row + kk);
        const v2f bb2 = *(const v2f*)(b2row + kk);
        const v2f a2  = a * a;
        acc = wmma_f32(a2, bb1, acc);   // -0.5 * sum x^2 / s^2
        acc = wmma_f32(a,  bb2, acc);   // + sum x*c / s^2
    }

    const int gcol = gn * 16 + row;     // N index for C/D layout
    const float cv = cst[gcol];
#pragma unroll
    for (int r = 0; r < 8; ++r) {
        const int m = r + 8 * khalf;    // D: vgpr r -> M=r (lanes 0-15), M=r+8 (16-31)
        G[(bm * 16 + m) * N_GAUSS + gcol] = __expf(acc[r] + cv);
    }
}

// ---------------------------------------------------------------------------
// Phase 2: out = G @ W.  M = BATCH, N = OUT_DIM, K = N_GAUSS.
// W is [N_GAUSS, OUT_DIM] row-major == K x N, so B elems stride by OUT_DIM.
// ---------------------------------------------------------------------------
__global__ void __launch_bounds__(128)
gkan_out(const float* __restrict__ G,
         const float* __restrict__ W,
         float* __restrict__ out) {
    const int wave  = (blockIdx.x * blockDim.x + threadIdx.x) >> 5;
    const int lane  = threadIdx.x & 31;
    const int on    = wave & ((OUT_DIM / 16) - 1);  // out tile (0..15)
    const int bm    = wave >> 4;                    // batch tile (0..63)
    const int row   = lane & 15;
    const int khalf = lane >> 4;

    const float* grow = G + (bm * 16 + row) * N_GAUSS;
    const float* wcol = W + on * 16 + row;          // fixed N, stride OUT_DIM in K

    v8f acc = {};
#pragma unroll 4
    for (int k0 = 0; k0 < N_GAUSS; k0 += 4) {
        const int kk = k0 + 2 * khalf;
        const v2f a = *(const v2f*)(grow + kk);
        v2f b;
        b.x = wcol[kk * OUT_DIM];
        b.y = wcol[(kk + 1) * OUT_DIM];
        acc = wmma_f32(a, b, acc);
    }

#pragma unroll
    for (int r = 0; r < 8; ++r) {
        const int m = r + 8 * khalf;
        out[(bm * 16 + m) * OUT_DIM + on * 16 + row] = acc[r];
    }
}

// ---------------------------------------------------------------------------
extern "C" void kernel_launch(void* const* d_in, const int* in_sizes, int n_in,
                              void* d_out, int out_size, void* d_ws, size_t ws_size,
                              hipStream_t stream) {
    (void)in_sizes; (void)n_in; (void)out_size; (void)ws_size;

    const float* x       = (const float*)d_in[0];
    const float* centers = (const float*)d_in[1];
    const float* stds    = (const float*)d_in[2];
    const float* weights = (const float*)d_in[3];
    float*       out     = (float*)d_out;

    // Workspace layout (floats): b1p | b2 | cst | G   (~3.1 MB total)
    float* ws  = (float*)d_ws;
    float* b1p = ws;                                   // 512*256
    float* b2  = b1p + N_GAUSS * IN_DIM;               // 512*256
    float* cst = b2  + N_GAUSS * IN_DIM;               // 512
    float* G   = cst + 512;                            // 1024*512 (16B aligned)

    // 1) prep tables + per-gaussian constant
    gkan_prep<<<N_GAUSS, IN_DIM, 0, stream>>>(centers, stds, b1p, b2, cst);

    // 2) fused distance GEMMs + exp -> G  (2048 tiles, 4 waves/block)
    const int tilesB = (BATCH / 16) * (N_GAUSS / 16);  // 2048
    gkan_gauss<<<tilesB / 4, 128, 0, stream>>>(x, b1p, b2, cst, G);

    // 3) out = G @ W  (1024 tiles, 4 waves/block)
    const int tilesC = (BATCH / 16) * (OUT_DIM / 16);  // 1024
    gkan_out<<<tilesC / 4, 128, 0, stream>>>(G, weights, out);
}